// GIN_50981261804329
// MI455X (gfx1250) — compile-verified
//
#include <hip/hip_runtime.h>

typedef __attribute__((ext_vector_type(2))) float v2f;
typedef __attribute__((ext_vector_type(8))) float v8f;

#define DNODE 64

// ---------------------------------------------------------------------------
// Zero a float buffer
// ---------------------------------------------------------------------------
__global__ void k_zero(float* __restrict__ p, int n) {
    int i = blockIdx.x * blockDim.x + threadIdx.x;
    if (i < n) p[i] = 0.0f;
}

// ---------------------------------------------------------------------------
// Edge scatter: agg[dst[e]] += h[src[e]]  (16 threads per edge, float4 each)
// ---------------------------------------------------------------------------
__global__ void k_scatter(const float* __restrict__ h,
                          const int* __restrict__ src,
                          const int* __restrict__ dst,
                          float* __restrict__ agg, int E) {
    long long idx = (long long)blockIdx.x * blockDim.x + threadIdx.x;
    int e  = (int)(idx >> 4);
    int c4 = (int)(idx & 15);
    if (e >= E) return;
    int s = src[e];
    int d = dst[e];
    float4 v = ((const float4*)(h + (size_t)s * DNODE))[c4];
    float* p = agg + (size_t)d * DNODE + c4 * 4;
    __hip_atomic_fetch_add(p + 0, v.x, __ATOMIC_RELAXED, __HIP_MEMORY_SCOPE_AGENT);
    __hip_atomic_fetch_add(p + 1, v.y, __ATOMIC_RELAXED, __HIP_MEMORY_SCOPE_AGENT);
    __hip_atomic_fetch_add(p + 2, v.z, __ATOMIC_RELAXED, __HIP_MEMORY_SCOPE_AGENT);
    __hip_atomic_fetch_add(p + 3, v.w, __ATOMIC_RELAXED, __HIP_MEMORY_SCOPE_AGENT);
}

// ---------------------------------------------------------------------------
// Fused GIN MLP:  h_out = relu((h_in + agg) @ Wa + ba) @ Wb + bb
// Block = 128 threads = 4 waves; each wave owns 16 rows.
// fp32 WMMA 16x16x4: A 16x4 (lane<16: M=lane,K={4s,4s+1}; lane>=16: M=lane-16,
// K={4s+2,4s+3}); B 4x16 mirrored; C/D 16x16 (vgpr r: lanes0-15 M=r, 16-31 M=r+8).
// Weights staged into LDS pair-interleaved: sW[p*64+col] = (W[2p][col], W[2p+1][col])
// so each B operand is one aligned ds_load_b64 into an even VGPR pair.
// ---------------------------------------------------------------------------
__global__ void __launch_bounds__(128)
k_gin_mlp(const float* __restrict__ h_in, const float* __restrict__ agg,
          const float* __restrict__ Wa, const float* __restrict__ ba,
          const float* __restrict__ Wb, const float* __restrict__ bb,
          float* __restrict__ h_out, int N) {
    __shared__ __align__(16) v2f sWa[32 * DNODE];   // pair-interleaved Wa
    __shared__ __align__(16) v2f sWb[32 * DNODE];   // pair-interleaved Wb
    __shared__ float sBa[DNODE];
    __shared__ float sBb[DNODE];
    __shared__ __align__(16) float sRT[4][16 * DNODE];  // r tile -> t tile -> d tile

    const int tid  = threadIdx.x;
    const int wid  = tid >> 5;
    const int lane = tid & 31;
    const int half = lane >> 4;   // 0: K pair {4s,4s+1}, 1: K pair {4s+2,4s+3}
    const int m    = lane & 15;   // A row within tile / B,C column within tile
    const int rowbase = blockIdx.x * 64 + wid * 16;

    // --- stage weights (pair-interleaved) + biases to LDS (whole block) ---
    for (int i = tid; i < 32 * DNODE; i += 128) {
        int p = i >> 6;   // k-pair index 0..31
        int c = i & 63;   // column
        v2f w;
        w.x = Wa[(2 * p) * DNODE + c];
        w.y = Wa[(2 * p + 1) * DNODE + c];
        sWa[i] = w;
    }
    for (int i = tid; i < 32 * DNODE; i += 128) {
        int p = i >> 6;
        int c = i & 63;
        v2f w;
        w.x = Wb[(2 * p) * DNODE + c];
        w.y = Wb[(2 * p + 1) * DNODE + c];
        sWb[i] = w;
    }
    if (tid < DNODE) { sBa[tid] = ba[tid]; sBb[tid] = bb[tid]; }

    // --- stage r = h_in + agg for this wave's 16 rows (coalesced float4) ---
    for (int i = lane; i < 16 * (DNODE / 4); i += 32) {
        int r  = i >> 4;   // row 0..15
        int c4 = i & 15;   // float4 column
        int grow = rowbase + r;
        grow = (grow < N) ? grow : (N - 1);
        float4 hv = ((const float4*)(h_in + (size_t)grow * DNODE))[c4];
        float4 av = ((const float4*)(agg  + (size_t)grow * DNODE))[c4];
        float4 rv;
        rv.x = hv.x + av.x; rv.y = hv.y + av.y;
        rv.z = hv.z + av.z; rv.w = hv.w + av.w;
        ((float4*)(sRT[wid]))[i] = rv;
    }
    __syncthreads();

    // --- load A operands (r tile) into registers: 16 k-steps ---
    v2f a[16];
#pragma unroll
    for (int s = 0; s < 16; ++s) {
        int k = 4 * s + 2 * half;
        a[s] = *(const v2f*)&sRT[wid][m * DNODE + k];
    }
    __syncthreads();   // all reads of sRT done before it is reused as t-tile

    // --- GEMM1: t = relu(r @ Wa + ba) -> sRT ---
#pragma unroll
    for (int nt = 0; nt < 4; ++nt) {
        v8f acc = {};
        const int col = nt * 16 + m;
#pragma unroll
        for (int s = 0; s < 16; ++s) {
            int p = 2 * s + half;            // k-pair index
            v2f b = sWa[p * DNODE + col];    // single ds_load_b64
            acc = __builtin_amdgcn_wmma_f32_16x16x4_f32(
                false, a[s], false, b, (short)0, acc, false, false);
        }
        float bias = sBa[col];
#pragma unroll
        for (int r = 0; r < 8; ++r) {
            float v = acc[r] + bias;
            v = v > 0.0f ? v : 0.0f;
            sRT[wid][(r + 8 * half) * DNODE + col] = v;
        }
    }
    __syncthreads();

    // --- load A operands (t tile) ---
    v2f a2[16];
#pragma unroll
    for (int s = 0; s < 16; ++s) {
        int k = 4 * s + 2 * half;
        a2[s] = *(const v2f*)&sRT[wid][m * DNODE + k];
    }
    __syncthreads();   // t reads done; sRT now reused for the output d-tile

    // --- GEMM2: d = t @ Wb + bb -> sRT (then coalesced store) ---
#pragma unroll
    for (int nt = 0; nt < 4; ++nt) {
        v8f acc = {};
        const int col = nt * 16 + m;
#pragma unroll
        for (int s = 0; s < 16; ++s) {
            int p = 2 * s + half;
            v2f b = sWb[p * DNODE + col];
            acc = __builtin_amdgcn_wmma_f32_16x16x4_f32(
                false, a2[s], false, b, (short)0, acc, false, false);
        }
        float bias = sBb[col];
#pragma unroll
        for (int r = 0; r < 8; ++r)
            sRT[wid][(r + 8 * half) * DNODE + col] = acc[r] + bias;
    }
    __syncthreads();

    // --- coalesced float4 store of the 16x64 output tile ---
    for (int i = lane; i < 16 * (DNODE / 4); i += 32) {
        int r  = i >> 4;
        int c4 = i & 15;
        int grow = rowbase + r;
        if (grow < N)
            ((float4*)(h_out + (size_t)grow * DNODE))[c4] = ((const float4*)(sRT[wid]))[i];
    }
}

// ---------------------------------------------------------------------------
// Column sums of final h: colsum[c] = sum_n h[n][c]   (one block per column)
// ---------------------------------------------------------------------------
__global__ void k_colsum(const float* __restrict__ h, float* __restrict__ colsum, int N) {
    __shared__ float red[256];
    int col = blockIdx.x;
    float s = 0.0f;
    for (int row = threadIdx.x; row < N; row += 256)
        s += h[(size_t)row * DNODE + col];
    red[threadIdx.x] = s;
    __syncthreads();
    for (int off = 128; off > 0; off >>= 1) {
        if (threadIdx.x < off) red[threadIdx.x] += red[threadIdx.x + off];
        __syncthreads();
    }
    if (threadIdx.x == 0) colsum[col] = red[0];
}

// ---------------------------------------------------------------------------
// Head: out[j] = (colsum/N) @ Wout + bout, j = 0..15
// ---------------------------------------------------------------------------
__global__ void k_head(const float* __restrict__ colsum, const float* __restrict__ Wout,
                       const float* __restrict__ bout, float* __restrict__ out, float invN) {
    int j = threadIdx.x;
    if (j >= 16) return;
    float s = 0.0f;
    for (int c = 0; c < DNODE; ++c)
        s += colsum[c] * Wout[c * 16 + j];
    out[j] = s * invN + bout[j];
}

// ---------------------------------------------------------------------------
extern "C" void kernel_launch(void* const* d_in, const int* in_sizes, int n_in,
                              void* d_out, int out_size, void* d_ws, size_t ws_size,
                              hipStream_t stream) {
    const float* features = (const float*)d_in[0];
    const int*   src      = (const int*)d_in[1];
    const int*   dst      = (const int*)d_in[2];
    const float* Wl[3][2] = {
        {(const float*)d_in[3],  (const float*)d_in[5]},
        {(const float*)d_in[7],  (const float*)d_in[9]},
        {(const float*)d_in[11], (const float*)d_in[13]},
    };
    const float* Bl[3][2] = {
        {(const float*)d_in[4],  (const float*)d_in[6]},
        {(const float*)d_in[8],  (const float*)d_in[10]},
        {(const float*)d_in[12], (const float*)d_in[14]},
    };
    const float* Wout = (const float*)d_in[15];
    const float* bout = (const float*)d_in[16];
    float* out = (float*)d_out;

    const int N = in_sizes[0] / DNODE;
    const int E = in_sizes[1];

    float* ws     = (float*)d_ws;
    float* agg    = ws;
    float* hA     = agg + (size_t)N * DNODE;
    float* hB     = hA  + (size_t)N * DNODE;
    float* colsum = hB  + (size_t)N * DNODE;

    float* bufs[2] = {hA, hB};
    const float* hin = features;

    const int nFeat = N * DNODE;
    for (int l = 0; l < 3; ++l) {
        k_zero<<<(nFeat + 255) / 256, 256, 0, stream>>>(agg, nFeat);
        long long nScat = (long long)E * 16;
        k_scatter<<<(unsigned)((nScat + 255) / 256), 256, 0, stream>>>(hin, src, dst, agg, E);
        float* hout = bufs[l & 1];
        k_gin_mlp<<<(N + 63) / 64, 128, 0, stream>>>(
            hin, agg, Wl[l][0], Bl[l][0], Wl[l][1], Bl[l][1], hout, N);
        hin = hout;
    }
    k_colsum<<<64, 256, 0, stream>>>(hin, colsum, N);
    k_head<<<1, 32, 0, stream>>>(colsum, Wout, bout, out, 1.0f / (float)N);
}